// pygGATModel_28922309771524
// MI455X (gfx1250) — compile-verified
//
#include <hip/hip_runtime.h>
#include <math.h>

#define N_NODES   100000
#define N_EDGES   1600000
#define E_TOT     (N_EDGES + N_NODES)   // reference adds self-loops
#define DIM       64
#define NCLS      10
#define NEG_SLOPE 0.2f

typedef float v8f __attribute__((ext_vector_type(8)));
typedef float v2f __attribute__((ext_vector_type(2)));

// ---------------------------------------------------------------------------
// GEMM: H[N x 64] = A[N x 64] * W[64 x 64], one wave per 16x16 output tile,
// f32 WMMA (V_WMMA_F32_16X16X4_F32), K stepped by 4, 16 wmma per tile.
// Lane layouts per CDNA5 ISA 7.12.2:
//   A 16x4 : lanes 0-15 M=0..15 K={k0,k0+1}; lanes 16-31 K={k0+2,k0+3}
//   B 4x16 : VGPR0 K=k0(+2 for hi lanes), VGPR1 K=k0+1(+2), N = lane&15
//   C/D    : VGPR v -> M = v (lanes 0-15) / v+8 (lanes 16-31), N = lane&15
// ---------------------------------------------------------------------------
__global__ __launch_bounds__(128) void gemm64_wmma(const float* __restrict__ A,
                                                   const float* __restrict__ W,
                                                   float* __restrict__ H) {
    int wave = (int)((blockIdx.x * blockDim.x + threadIdx.x) >> 5);
    int lane = threadIdx.x & 31;
    int m0 = (wave >> 2) << 4;          // node-row tile
    int n0 = (wave & 3) << 4;           // output-column tile
    if (m0 >= N_NODES) return;

    int l16 = lane & 15;
    int kb  = (lane >> 4) << 1;         // 0 for lanes 0-15, 2 for lanes 16-31
    int row = m0 + l16;
    int col = n0 + l16;

    const float* arow = A + (size_t)row * DIM + kb;   // A[row, kb + k0 ..]
    const float* wcol = W + (size_t)kb * DIM + col;   // W[kb + k0, col]

    v8f c = {};
#pragma unroll
    for (int k0 = 0; k0 < DIM; k0 += 4) {
        v2f a = *(const v2f*)(arow + k0);             // A[row,k0+kb], A[row,k0+kb+1]
        v2f b;
        b.x = wcol[(size_t)k0 * DIM];                 // W[k0+kb,   col]
        b.y = wcol[(size_t)(k0 + 1) * DIM];           // W[k0+kb+1, col]
        c = __builtin_amdgcn_wmma_f32_16x16x4_f32(false, a, false, b,
                                                  (short)0, c, false, false);
    }

    int mo = (lane >> 4) << 3;                        // 0 or 8
#pragma unroll
    for (int v = 0; v < 8; ++v)
        H[(size_t)(m0 + mo + v) * DIM + n0 + l16] = c[v];
}

// ---------------------------------------------------------------------------
// Per-node attention logits: as[n] = h[n].a_src, ad[n] = h[n].a_dst
// One wave32 per node, 2 elements per lane, xor-shuffle tree reduction.
// ---------------------------------------------------------------------------
__global__ __launch_bounds__(256) void alpha_kernel(const float* __restrict__ H,
                                                    const float* __restrict__ a_src,
                                                    const float* __restrict__ a_dst,
                                                    float* __restrict__ as,
                                                    float* __restrict__ ad) {
    int wave = (int)((blockIdx.x * blockDim.x + threadIdx.x) >> 5);
    int lane = threadIdx.x & 31;
    if (wave >= N_NODES) return;
    const float* h = H + (size_t)wave * DIM;
    float h0 = h[lane], h1 = h[lane + 32];
    float s = h0 * a_src[lane] + h1 * a_src[lane + 32];
    float d = h0 * a_dst[lane] + h1 * a_dst[lane + 32];
#pragma unroll
    for (int off = 16; off > 0; off >>= 1) {
        s += __shfl_xor(s, off, 32);
        d += __shfl_xor(d, off, 32);
    }
    if (lane == 0) { as[wave] = s; ad[wave] = d; }
}

// Monotonic float<->uint mapping so integer atomicMax implements float max.
__device__ __forceinline__ unsigned f2mono(float f) {
    unsigned u = __float_as_uint(f);
    return (u & 0x80000000u) ? ~u : (u | 0x80000000u);
}
__device__ __forceinline__ float mono2f(unsigned u) {
    return __uint_as_float((u & 0x80000000u) ? (u ^ 0x80000000u) : ~u);
}

__global__ void init_kernel(unsigned* __restrict__ emax_u,
                            float* __restrict__ denom,
                            float* __restrict__ agg,
                            float* __restrict__ pooled) {
    size_t t = (size_t)blockIdx.x * blockDim.x + threadIdx.x;
    const size_t total = (size_t)N_NODES * DIM;
    if (t < total)   agg[t] = 0.0f;
    if (t < N_NODES) { emax_u[t] = 0u; denom[t] = 0.0f; }  // 0 maps below any real
    if (t < DIM)     pooled[t] = 0.0f;
}

__global__ void edge_score_max(const int* __restrict__ ei,
                               const float* __restrict__ as,
                               const float* __restrict__ ad,
                               float* __restrict__ escore,
                               unsigned* __restrict__ emax_u) {
    int e = (int)(blockIdx.x * blockDim.x + threadIdx.x);
    if (e >= E_TOT) return;
    int src, dst;
    if (e < N_EDGES) { src = ei[e]; dst = ei[N_EDGES + e]; }
    else             { src = dst = e - N_EDGES; }          // self-loop
    float v = as[src] + ad[dst];
    v = (v > 0.0f) ? v : NEG_SLOPE * v;                    // LeakyReLU
    escore[e] = v;
    atomicMax(emax_u + dst, f2mono(v));
}

__global__ void edge_exp_sum(const int* __restrict__ ei,
                             const unsigned* __restrict__ emax_u,
                             float* __restrict__ escore,
                             float* __restrict__ denom) {
    int e = (int)(blockIdx.x * blockDim.x + threadIdx.x);
    if (e >= E_TOT) return;
    int dst = (e < N_EDGES) ? ei[N_EDGES + e] : (e - N_EDGES);
    float ex = __expf(escore[e] - mono2f(emax_u[dst]));
    escore[e] = ex;
    atomicAdd(denom + dst, ex);
}

// One wave32 per edge: coalesced gather of h[src] (L2-resident, 25.6 MB),
// scaled scatter-add into agg[dst] (2 f32 atomics per lane).
__global__ __launch_bounds__(256) void aggregate(const int* __restrict__ ei,
                                                 const float* __restrict__ ex,
                                                 const float* __restrict__ denom,
                                                 const float* __restrict__ H,
                                                 float* __restrict__ agg) {
    int wave = (int)((blockIdx.x * blockDim.x + threadIdx.x) >> 5);
    int lane = threadIdx.x & 31;
    if (wave >= E_TOT) return;
    int src, dst;
    if (wave < N_EDGES) { src = ei[wave]; dst = ei[N_EDGES + wave]; }
    else                { src = dst = wave - N_EDGES; }
    float attn = ex[wave] / denom[dst];
    const float* hs = H + (size_t)src * DIM;
    float*       od = agg + (size_t)dst * DIM;
    atomicAdd(od + lane,      attn * hs[lane]);
    atomicAdd(od + lane + 32, attn * hs[lane + 32]);
}

__global__ void bias_relu(float* __restrict__ agg, const float* __restrict__ b) {
    size_t t = (size_t)blockIdx.x * blockDim.x + threadIdx.x;
    if (t >= (size_t)N_NODES * DIM) return;
    agg[t] = fmaxf(agg[t] + b[t & 63], 0.0f);
}

// Column sums for mean pooling: 4 rows x 64 cols per block, LDS reduce,
// one atomicAdd per column per block.
__global__ __launch_bounds__(256) void col_sum(const float* __restrict__ H,
                                               float* __restrict__ pooled) {
    __shared__ float sm[256];
    int col = threadIdx.x & 63;
    int r   = threadIdx.x >> 6;           // 0..3
    float s = 0.0f;
    for (int row = blockIdx.x * 4 + r; row < N_NODES; row += gridDim.x * 4)
        s += H[(size_t)row * DIM + col];
    sm[threadIdx.x] = s;
    __syncthreads();
    if (threadIdx.x < 64) {
        s = sm[threadIdx.x] + sm[threadIdx.x + 64] +
            sm[threadIdx.x + 128] + sm[threadIdx.x + 192];
        atomicAdd(pooled + col, s);
    }
}

__global__ void final_kernel(const float* __restrict__ pooled,
                             const float* __restrict__ Wc,
                             const float* __restrict__ bc,
                             float* __restrict__ out) {
    int c = threadIdx.x;
    if (c >= NCLS) return;
    const float inv = 1.0f / (float)N_NODES;
    float s = bc[c];
#pragma unroll
    for (int k = 0; k < DIM; ++k)
        s += pooled[k] * inv * Wc[k * NCLS + c];
    out[c] = s;
}

// ---------------------------------------------------------------------------
extern "C" void kernel_launch(void* const* d_in, const int* in_sizes, int n_in,
                              void* d_out, int out_size, void* d_ws, size_t ws_size,
                              hipStream_t stream) {
    const float* x   = (const float*)d_in[0];
    const int*   ei  = (const int*)d_in[1];     // (2, E) row-major, int32 per harness
    const float* W1  = (const float*)d_in[2];
    const float* a1s = (const float*)d_in[3];
    const float* a1d = (const float*)d_in[4];
    const float* b1  = (const float*)d_in[5];
    const float* W2  = (const float*)d_in[6];
    const float* a2s = (const float*)d_in[7];
    const float* a2d = (const float*)d_in[8];
    const float* b2  = (const float*)d_in[9];
    const float* Wc  = (const float*)d_in[10];
    const float* bc  = (const float*)d_in[11];
    float* out = (float*)d_out;

    char* ws = (char*)d_ws;
    size_t off = 0;
    auto alloc = [&](size_t bytes) -> char* {
        char* p = ws + off;
        off += (bytes + 255) & ~(size_t)255;
        return p;
    };
    const size_t NHb = (size_t)N_NODES * DIM * sizeof(float);   // 25.6 MB
    float*    h_tmp   = (float*)alloc(NHb);
    float*    agg     = (float*)alloc(NHb);
    float*    alpha_s = (float*)alloc((size_t)N_NODES * sizeof(float));
    float*    alpha_d = (float*)alloc((size_t)N_NODES * sizeof(float));
    unsigned* emax_u  = (unsigned*)alloc((size_t)N_NODES * sizeof(unsigned));
    float*    denom   = (float*)alloc((size_t)N_NODES * sizeof(float));
    float*    escore  = (float*)alloc((size_t)E_TOT * sizeof(float));
    float*    pooled  = (float*)alloc(DIM * sizeof(float));
    if (off > ws_size) return;                  // deterministic guard

    const int gemm_blocks = (N_NODES / 16) * (DIM / 16) / 4;            // 6250
    const int elem_blocks = (N_NODES * DIM + 255) / 256;                // 25000
    const int edge_blocks = (E_TOT + 255) / 256;
    const int aggw_blocks = (int)(((size_t)E_TOT * 32 + 255) / 256);    // 212500

    auto gat_layer = [&](const float* X, const float* W,
                         const float* aw_s, const float* aw_d, const float* b) {
        gemm64_wmma   <<<gemm_blocks, 128, 0, stream>>>(X, W, h_tmp);
        alpha_kernel  <<<(N_NODES * 32 + 255) / 256, 256, 0, stream>>>(h_tmp, aw_s, aw_d,
                                                                       alpha_s, alpha_d);
        init_kernel   <<<elem_blocks, 256, 0, stream>>>(emax_u, denom, agg, pooled);
        edge_score_max<<<edge_blocks, 256, 0, stream>>>(ei, alpha_s, alpha_d, escore, emax_u);
        edge_exp_sum  <<<edge_blocks, 256, 0, stream>>>(ei, emax_u, escore, denom);
        aggregate     <<<aggw_blocks, 256, 0, stream>>>(ei, escore, denom, h_tmp, agg);
        bias_relu     <<<elem_blocks, 256, 0, stream>>>(agg, b);
    };

    gat_layer(x,   W1, a1s, a1d, b1);   // layer 1: agg = relu(conv1(x))
    gat_layer(agg, W2, a2s, a2d, b2);   // layer 2: agg = relu(conv2(agg))

    col_sum     <<<256, 256, 0, stream>>>(agg, pooled);
    final_kernel<<<1, 32, 0, stream>>>(pooled, Wc, bc, out);
}